// SageEncoder_20100446946058
// MI455X (gfx1250) — compile-verified
//
#include <hip/hip_runtime.h>

typedef float v2f __attribute__((ext_vector_type(2)));
typedef float v8f __attribute__((ext_vector_type(8)));

#define DFEAT  128
#define WPITCH 130   // LDS row pitch in floats: stride 130 dwords -> conflict-free b64 reads
#define EPSN   1e-12f

// ---------------------------------------------------------------------------
// Mean-aggregation scatter: one wave32 per edge. Lane l moves 4 consecutive
// floats (float4 gather of x[src] row, 4x global_atomic_add_f32 into agg[dst]).
// ---------------------------------------------------------------------------
__global__ void __launch_bounds__(256) sage_scatter(
    const float* __restrict__ x, const int* __restrict__ src, const int* __restrict__ dst,
    float* __restrict__ agg, float* __restrict__ cnt, int nEdges)
{
    const int lane   = threadIdx.x & 31;
    const int wave   = (blockIdx.x * blockDim.x + threadIdx.x) >> 5;
    const int nWaves = (gridDim.x * blockDim.x) >> 5;
    for (int e = wave; e < nEdges; e += nWaves) {
        const int s = src[e];
        const int d = dst[e];
        const float4 v = ((const float4*)(x + (size_t)s * DFEAT))[lane];
        float* a = agg + (size_t)d * DFEAT + lane * 4;
        atomicAdd(a + 0, v.x);
        atomicAdd(a + 1, v.y);
        atomicAdd(a + 2, v.z);
        atomicAdd(a + 3, v.w);
        if (lane == 0) atomicAdd(cnt + d, 1.0f);
    }
}

// ---------------------------------------------------------------------------
// Fused SAGE layer: out = rownorm( (agg/cnt) @ Wl^T + b + x @ Wr^T ), opt. ReLU.
// One wave per 16-row block; 8 column tiles of 16 -> full 16x128 output block.
// fp32 WMMA 16x16x4; K loop of 32 steps, 16 v_wmma per step (two GEMMs fused).
// All 16 B tiles for a k-step are loaded into registers BEFORE the WMMA burst
// so the scheduler can overlap DS loads with the matrix pipe (partial dscnt
// waits instead of dscnt==0 stalls).
// ---------------------------------------------------------------------------
__global__ void __launch_bounds__(256) sage_gemm(
    const float* __restrict__ agg, const float* __restrict__ cnt,
    const float* __restrict__ xin,
    const float* __restrict__ Wl, const float* __restrict__ Wr,
    const float* __restrict__ bias,
    float* __restrict__ out, int nRows, int doRelu)
{
    extern __shared__ float lds[];
    float* sWl = lds;                   // sW[n*WPITCH + k] = W[n][k]  (B = W^T)
    float* sWr = lds + WPITCH * DFEAT;

    for (int i = threadIdx.x; i < (DFEAT * DFEAT) / 4; i += blockDim.x) {
        const int n = (i * 4) / DFEAT;
        const int k = (i * 4) % DFEAT;
        const float4 wl4 = ((const float4*)Wl)[i];
        const float4 wr4 = ((const float4*)Wr)[i];
        float* pl = sWl + n * WPITCH + k;
        pl[0] = wl4.x; pl[1] = wl4.y; pl[2] = wl4.z; pl[3] = wl4.w;
        float* pr = sWr + n * WPITCH + k;
        pr[0] = wr4.x; pr[1] = wr4.y; pr[2] = wr4.z; pr[3] = wr4.w;
    }
    __syncthreads();

    const int lane   = threadIdx.x & 31;
    const int lm     = lane & 15;        // column-in-tile / row-in-A-tile
    const int hi     = lane >> 4;        // half-wave select
    const int wave   = (blockIdx.x * blockDim.x + threadIdx.x) >> 5;
    const int nWaves = (gridDim.x * blockDim.x) >> 5;
    const int nBlk   = nRows >> 4;

    // Per-lane base into the B-tile LDS region: column n = cg*16 + lm, k-pair at 2*hi.
    const float* bBasL = sWl + (size_t)lm * WPITCH;
    const float* bBasR = sWr + (size_t)lm * WPITCH;

    for (int rb = wave; rb < nBlk; rb += nWaves) {
        const int row0 = rb * 16;
        const int r    = row0 + lm;
        const float rc = 1.0f / fmaxf(cnt[r], 1.0f);   // mean = agg / max(cnt,1)

        const float* aRow = agg + (size_t)r * DFEAT;
        const float* xRow = xin + (size_t)r * DFEAT;

        v8f acc[8];
        #pragma unroll
        for (int cg = 0; cg < 8; ++cg) acc[cg] = (v8f){0,0,0,0,0,0,0,0};

        for (int k0 = 0; k0 < DFEAT; k0 += 4) {
            const int k = k0 + 2 * hi;   // ISA A-tile: VGPR0 holds K=k0(+2), VGPR1 K=k0+1(+3)
            const float2 am = *(const float2*)(aRow + k);
            const float2 ax = *(const float2*)(xRow + k);
            v2f aMean; aMean[0] = am.x * rc; aMean[1] = am.y * rc;
            v2f aX;    aX[0]    = ax.x;      aX[1]    = ax.y;

            // Batch-load all 16 B tiles for this k-step first...
            v2f bl[8], br[8];
            #pragma unroll
            for (int cg = 0; cg < 8; ++cg) {
                bl[cg] = *(const v2f*)(bBasL + cg * (16 * WPITCH) + k);
                br[cg] = *(const v2f*)(bBasR + cg * (16 * WPITCH) + k);
            }
            // ...then burst the 16 WMMAs (independent accumulators -> no stalls).
            #pragma unroll
            for (int cg = 0; cg < 8; ++cg) {
                acc[cg] = __builtin_amdgcn_wmma_f32_16x16x4_f32(
                    false, aMean, false, bl[cg], (short)0, acc[cg], false, false);
                acc[cg] = __builtin_amdgcn_wmma_f32_16x16x4_f32(
                    false, aX, false, br[cg], (short)0, acc[cg], false, false);
            }
        }

        // Epilogue: +bias, L2 row norm (butterfly over the 16-lane half), ReLU, store.
        float ssum[8];
        #pragma unroll
        for (int v = 0; v < 8; ++v) ssum[v] = 0.0f;
        #pragma unroll
        for (int cg = 0; cg < 8; ++cg) {
            const float bv = bias[cg * 16 + lm];
            #pragma unroll
            for (int v = 0; v < 8; ++v) {
                const float t = acc[cg][v] + bv;
                acc[cg][v] = t;
                ssum[v] += t * t;
            }
        }
        #pragma unroll
        for (int m = 1; m < 16; m <<= 1) {
            #pragma unroll
            for (int v = 0; v < 8; ++v) ssum[v] += __shfl_xor(ssum[v], m, 32);
        }
        float rn[8];
        #pragma unroll
        for (int v = 0; v < 8; ++v) rn[v] = 1.0f / fmaxf(sqrtf(ssum[v]), EPSN);

        #pragma unroll
        for (int cg = 0; cg < 8; ++cg) {
            #pragma unroll
            for (int v = 0; v < 8; ++v) {
                float val = acc[cg][v] * rn[v];
                if (doRelu) val = fmaxf(val, 0.0f);
                out[(size_t)(row0 + v + 8 * hi) * DFEAT + cg * 16 + lm] = val;
            }
        }
    }
}

// ---------------------------------------------------------------------------
extern "C" void kernel_launch(void* const* d_in, const int* in_sizes, int n_in,
                              void* d_out, int out_size, void* d_ws, size_t ws_size,
                              hipStream_t stream)
{
    const float* x   = (const float*)d_in[0];
    const int*   ei  = (const int*)d_in[1];
    const float* Wl1 = (const float*)d_in[2];
    const float* b1  = (const float*)d_in[3];
    const float* Wr1 = (const float*)d_in[4];
    const float* Wl2 = (const float*)d_in[5];
    const float* b2  = (const float*)d_in[6];
    const float* Wr2 = (const float*)d_in[7];
    float* out = (float*)d_out;

    const int nNodes = in_sizes[0] / DFEAT;   // 50000
    const int nEdges = in_sizes[1] / 2;       // 640000
    const int* src = ei;
    const int* dst = ei + nEdges;

    float* agg = (float*)d_ws;                         // N*D
    float* cnt = agg + (size_t)nNodes * DFEAT;         // N
    float* h   = cnt + nNodes;                         // N*D

    const size_t shmem = (size_t)2 * WPITCH * DFEAT * sizeof(float); // 133 KB (<320 KB/WGP)
    const size_t zBytes = ((size_t)nNodes * DFEAT + nNodes) * sizeof(float);

    const int scatterBlocks = (nEdges + 7) / 8;   // 8 waves/block, one edge per wave
    const int rowBlocks     = nNodes / 16;        // 3125
    const int gemmBlocks    = (rowBlocks + 7) / 8;

    // ---- layer 1: h = relu(rownorm(mean(x)@Wl1^T + b1 + x@Wr1^T)) ----
    hipMemsetAsync(agg, 0, zBytes, stream);
    sage_scatter<<<scatterBlocks, 256, 0, stream>>>(x, src, dst, agg, cnt, nEdges);
    sage_gemm<<<gemmBlocks, 256, shmem, stream>>>(agg, cnt, x, Wl1, Wr1, b1, h, nNodes, 1);

    // ---- layer 2: out = rownorm(mean(h)@Wl2^T + b2 + h@Wr2^T) ----
    hipMemsetAsync(agg, 0, zBytes, stream);
    sage_scatter<<<scatterBlocks, 256, 0, stream>>>(h, src, dst, agg, cnt, nEdges);
    sage_gemm<<<gemmBlocks, 256, shmem, stream>>>(agg, cnt, h, Wl2, Wr2, b2, out, nNodes, 0);
}